// incremental_HMM_8658654068931
// MI455X (gfx1250) — compile-verified
//
#include <hip/hip_runtime.h>

// CDNA5 / gfx1250 — wave32, WMMA f32 16x16x4.
typedef __attribute__((ext_vector_type(2))) float v2f;
typedef __attribute__((ext_vector_type(8))) float v8f;

#define LOG_2PI_F 1.8378770664093453f

// One wave (32 lanes) handles one timestep l.
//   A (16x4, M=r, K=4): lane<16 holds (a[r], b[l,r]) in K0/K1; lane>=16 holds (c[l,r], 0) in K2/K3,
//     with r = lane&15 (tile lo) or (lane&15)+16 (tile hi).  [ISA 16x4 f32 A layout]
//   B (4x16, K=4, N=f): lane<16 holds (x^2, x) for K0/K1; lane>=16 holds (1, 0) for K2/K3,
//     with f = fb + (lane&15).
//   D (16x16, M=r, N=f): VGPR p: lanes0-15 -> r=p, lanes16-31 -> r=p+8, f=lane&15.
// logit = a*x^2 + b*x + c, then logsumexp over r (16 in-lane values + shfl_xor(16) partner).
__global__ __launch_bounds__(256) void incremental_hmm_score_kernel(
    const float* __restrict__ X,         // [4096, 256]
    const float* __restrict__ h,         // [32]
    const float* __restrict__ mu_rates,  // [32]
    const float* __restrict__ sig,       // [32]
    float* __restrict__ out)             // [4096, 256]
{
    const int lane   = threadIdx.x & 31;
    const int wave   = threadIdx.x >> 5;
    const int l      = blockIdx.x * 8 + wave;     // 512 blocks * 8 waves = 4096
    const int r0     = lane & 15;
    const bool hiK   = lane >= 16;                // K=2/3 half of A and B

    // ---- per-l, per-component coefficients (computed once per wave) ----
    const float h_lo  = h[r0];
    const float h_hi  = h[r0 + 16];
    const float mu0lo = mu_rates[r0];
    const float mu0hi = mu_rates[r0 + 16];
    const float s_lo  = sig[r0];
    const float s_hi  = sig[r0 + 16];

    // total mixture weight: butterfly sum within each 16-lane group
    float hs = h_lo + h_hi;
    #pragma unroll
    for (int m = 1; m < 16; m <<= 1) hs += __shfl_xor(hs, m, 32);
    const float inv_hs = 1.0f / hs;

    const float shift = (float)(l / 5 + 1);       // cumulative mu increment

    const float invs2_lo = 1.0f / (s_lo * s_lo);
    const float invs2_hi = 1.0f / (s_hi * s_hi);
    const float a_lo = -0.5f * invs2_lo;
    const float a_hi = -0.5f * invs2_hi;
    const float base_lo = __logf(h_lo * inv_hs) - __logf(s_lo) - 0.5f * LOG_2PI_F;
    const float base_hi = __logf(h_hi * inv_hs) - __logf(s_hi) - 0.5f * LOG_2PI_F;
    const float mu_lo = mu0lo + shift;
    const float mu_hi = mu0hi + shift;
    const float b_lo = mu_lo * invs2_lo;
    const float b_hi = mu_hi * invs2_hi;
    const float c_lo = base_lo - 0.5f * mu_lo * mu_lo * invs2_lo;
    const float c_hi = base_hi - 0.5f * mu_hi * mu_hi * invs2_hi;

    // A matrices for r-tile lo (r=0..15) and hi (r=16..31)
    v2f A_lo, A_hi;
    A_lo.x = hiK ? c_lo : a_lo;   // K0 (lanes<16) / K2 (lanes>=16)
    A_lo.y = hiK ? 0.0f : b_lo;   // K1 / K3
    A_hi.x = hiK ? c_hi : a_hi;
    A_hi.y = hiK ? 0.0f : b_hi;

    const float* xrow = X   + (size_t)l * 256;
    float*       orow = out + (size_t)l * 256;

    #pragma unroll 4
    for (int fb = 0; fb < 256; fb += 16) {
        const float x = xrow[fb + r0];
        v2f B;
        B.x = hiK ? 1.0f : x * x;  // K0: x^2 pairs with a; K2: 1 pairs with c
        B.y = hiK ? 0.0f : x;      // K1: x   pairs with b; K3: 0

        v8f d0 = {};  // r = 0..15
        v8f d1 = {};  // r = 16..31
        d0 = __builtin_amdgcn_wmma_f32_16x16x4_f32(false, A_lo, false, B,
                                                   (short)0, d0, false, false);
        d1 = __builtin_amdgcn_wmma_f32_16x16x4_f32(false, A_hi, false, B,
                                                   (short)0, d1, false, false);

        // ---- logsumexp over r: 16 values in-lane, 16 in partner lane (xor 16) ----
        float m = d0[0];
        #pragma unroll
        for (int p = 1; p < 8; ++p) m = fmaxf(m, d0[p]);
        #pragma unroll
        for (int p = 0; p < 8; ++p) m = fmaxf(m, d1[p]);
        m = fmaxf(m, __shfl_xor(m, 16, 32));

        float ssum = 0.0f;
        #pragma unroll
        for (int p = 0; p < 8; ++p) ssum += __expf(d0[p] - m);
        #pragma unroll
        for (int p = 0; p < 8; ++p) ssum += __expf(d1[p] - m);
        ssum += __shfl_xor(ssum, 16, 32);

        const float res = m + __logf(ssum);

        // lanes 0..15 store f = fb + lane (coalesced 64B)
        if (!hiK) orow[fb + lane] = res;
    }
}

extern "C" void kernel_launch(void* const* d_in, const int* in_sizes, int n_in,
                              void* d_out, int out_size, void* d_ws, size_t ws_size,
                              hipStream_t stream) {
    (void)in_sizes; (void)n_in; (void)d_ws; (void)ws_size; (void)out_size;
    const float* X        = (const float*)d_in[0];  // [1, 4096, 256]
    const float* h        = (const float*)d_in[1];  // [1, 32]
    const float* mu_rates = (const float*)d_in[2];  // [1, 32]
    const float* sig      = (const float*)d_in[3];  // [1, 32]
    float* out            = (float*)d_out;          // [1, 4096, 256]

    dim3 grid(512);   // 512 blocks * 8 waves/block = 4096 timesteps
    dim3 block(256);  // 8 wave32 waves
    incremental_hmm_score_kernel<<<grid, block, 0, stream>>>(X, h, mu_rates, sig, out);
}